// FuzzyART_46643344835327
// MI455X (gfx1250) — compile-verified
//
#include <hip/hip_runtime.h>
#include <hip/hip_bf16.h>
#include <stdint.h>

// ---------------------------------------------------------------------------
// FuzzyART match on MI455X (gfx1250, wave32)
//   match[b,c] = sum_d min(x[b,d], cat[c,d]) / sum_d x[b,d]
//   score = match >= 0.75 ? match : 0 ; idx[b] = argmax_c score
// Plan: 64x64 output tile per 256-thread block; tiles staged to LDS by the
// Tensor Data Mover (pad_enable gives a 260-dword row stride) with an
// async-to-LDS fallback tier; denominators via V_WMMA_F32_16X16X4_F32
// against a ones B-matrix; min+add main loop (VALU-bound by roofline).
// ---------------------------------------------------------------------------

#define VIG 0.75f
#define DD 256              // feature dim (fixed by reference)
#define TB 64               // rows (b) per block
#define TC 64               // categories (c) per block
#define STRIDE 260          // LDS row stride in dwords (256 + 4 pad)
#define NTHREADS 256

#define XS_OFF 0
#define CS_OFF (TB * STRIDE)
#define RS_OFF (CS_OFF + TC * STRIDE)
#define SMEM_DWORDS (RS_OFF + TB)
#define SMEM_BYTES (SMEM_DWORDS * 4)

#if defined(__HIP_DEVICE_COMPILE__) && __has_builtin(__builtin_amdgcn_tensor_load_to_lds)
#define HAVE_TDM 1
#if __has_include(<hip/amd_detail/amd_gfx1250_TDM.h>)
#define TDM_ARGS6 1   // therock/clang-23 6-arg form
#endif
#endif

#if defined(__HIP_DEVICE_COMPILE__) && !defined(HAVE_TDM) && \
    __has_builtin(__builtin_amdgcn_global_load_async_to_lds_b128)
#define HAVE_ASYNC_LDS 1
#endif

#if defined(__HIP_DEVICE_COMPILE__) && __has_builtin(__builtin_amdgcn_wmma_f32_16x16x4_f32)
#define HAVE_WMMA_F32X4 1
#endif

// ---- stderr probes: which staging path did the device pass pick? ----------
#if defined(__HIP_DEVICE_COMPILE__)
#if defined(HAVE_TDM)
#warning "CDNA5-PATH: TENSOR_LOAD_TO_LDS (TDM) staging active"
#elif defined(HAVE_ASYNC_LDS)
#warning "CDNA5-PATH: GLOBAL_LOAD_ASYNC_TO_LDS_B128 staging active"
#else
#warning "CDNA5-PATH: plain vector staging fallback (no TDM/async builtins)"
#endif
#if defined(HAVE_WMMA_F32X4)
#warning "CDNA5-PATH: V_WMMA_F32_16X16X4_F32 row-sum active"
#else
#warning "CDNA5-PATH: scalar row-sum fallback (no wmma f32x4 builtin)"
#endif
#endif

typedef float v2f __attribute__((ext_vector_type(2)));
typedef float v8f __attribute__((ext_vector_type(8)));
typedef unsigned int u32x4 __attribute__((ext_vector_type(4)));
typedef int i32x4 __attribute__((ext_vector_type(4)));
typedef int i32x8 __attribute__((ext_vector_type(8)));

#if defined(HAVE_TDM)
// 2D TDM tile load: `rows` x 256 f32, row-major, padded 4 dwords per 256 in LDS.
__device__ __forceinline__ void tdm_load_tile_256(const float* gsrc,
                                                  unsigned lds_byte_off,
                                                  unsigned rows) {
  uint64_t ga = (uint64_t)(uintptr_t)gsrc;
  u32x4 g0;
  g0[0] = 1u;                                   // count=1 (user descriptor)
  g0[1] = lds_byte_off;                         // lds_addr
  g0[2] = (unsigned)ga;                         // global_addr[31:0]
  g0[3] = (unsigned)((ga >> 32) & 0x01FFFFFFull) | (2u << 30); // addr[56:32], type=2
  i32x8 g1;
  g1[0] = (2 << 16)        // data_size = 4B
        | (1 << 20)        // pad_enable
        | (7 << 22)        // pad_interval: 256 dwords
        | (3 << 25);       // pad_amount: 4 dwords
  g1[1] = (int)((unsigned)DD << 16);     // tensor_dim0 = 256
  g1[2] = (int)(rows << 16);             // tensor_dim1 = rows
  g1[3] = (int)((unsigned)DD << 16);     // tile_dim0 = 256
  g1[4] = (int)rows;                     // tile_dim1 = rows, tile_dim2 = 0
  g1[5] = DD;                            // tensor_dim0_stride = 256
  g1[6] = 0;
  g1[7] = 0;
  i32x4 gz = {0, 0, 0, 0};
#if defined(TDM_ARGS6)
  i32x8 gz8 = {0, 0, 0, 0, 0, 0, 0, 0};
  __builtin_amdgcn_tensor_load_to_lds(g0, g1, gz, gz, gz8, 0);
#else
  __builtin_amdgcn_tensor_load_to_lds(g0, g1, gz, gz, 0);
#endif
}
#endif

__global__ __launch_bounds__(NTHREADS)
void fuzzyart_match_kernel(const float* __restrict__ x,
                           const float* __restrict__ cats,
                           float* __restrict__ out, int B, int C) {
  extern __shared__ float smem[];
  const int tid = threadIdx.x;
  const int b0g = blockIdx.x * TB;
  const int c0g = blockIdx.y * TC;

  // ---- stage x-tile and cat-tile into LDS -------------------------------
#if defined(HAVE_TDM)
  if (tid == 0) {
    tdm_load_tile_256(x + (size_t)b0g * DD, (unsigned)(XS_OFF * 4), TB);
    tdm_load_tile_256(cats + (size_t)c0g * DD, (unsigned)(CS_OFF * 4), TC);
#if __has_builtin(__builtin_amdgcn_s_wait_tensorcnt)
    __builtin_amdgcn_s_wait_tensorcnt(0);
#endif
  }
  __syncthreads();
#elif defined(HAVE_ASYNC_LDS)
  {
    // Each lane moves 16B per op; 64 rows x 256 dwords per tile.
    typedef __attribute__((address_space(1))) const void as1c;
    typedef __attribute__((address_space(3))) void as3;
#pragma unroll
    for (int it = 0; it < (TB * (DD / 4)) / NTHREADS; ++it) {
      int i = tid + it * NTHREADS;
      int r = i >> 6;
      int c4 = (i & 63) << 2;
      __builtin_amdgcn_global_load_async_to_lds_b128(
          (as1c*)(x + (size_t)(b0g + r) * DD + c4),
          (as3*)&smem[XS_OFF + r * STRIDE + c4], 0, 0);
    }
#pragma unroll
    for (int it = 0; it < (TC * (DD / 4)) / NTHREADS; ++it) {
      int i = tid + it * NTHREADS;
      int r = i >> 6;
      int c4 = (i & 63) << 2;
      __builtin_amdgcn_global_load_async_to_lds_b128(
          (as1c*)(cats + (size_t)(c0g + r) * DD + c4),
          (as3*)&smem[CS_OFF + r * STRIDE + c4], 0, 0);
    }
#if __has_builtin(__builtin_amdgcn_s_wait_asynccnt)
    __builtin_amdgcn_s_wait_asynccnt(0);
#else
    asm volatile("s_wait_asynccnt 0" ::: "memory");
#endif
  }
  __syncthreads();
#else
  for (int i = tid; i < TB * (DD / 4); i += NTHREADS) {
    int r = i >> 6;
    int c4 = (i & 63) << 2;
    *(float4*)&smem[XS_OFF + r * STRIDE + c4] =
        *(const float4*)(x + (size_t)(b0g + r) * DD + c4);
  }
  for (int i = tid; i < TC * (DD / 4); i += NTHREADS) {
    int r = i >> 6;
    int c4 = (i & 63) << 2;
    *(float4*)&smem[CS_OFF + r * STRIDE + c4] =
        *(const float4*)(cats + (size_t)(c0g + r) * DD + c4);
  }
  __syncthreads();
#endif

  // ---- denominators: 64 row sums of the x-tile --------------------------
#if defined(HAVE_WMMA_F32X4)
  // Waves 0..3: row-group w = rows 16w..16w+15. B = ones 4x16 (layout-free),
  // so D[m,n] = sum_k A[m,k]; accumulate over K chunks of 4.
  if (tid < 128) {
    const int wave = tid >> 5;
    const int lane = tid & 31;
    const int row = wave * 16 + (lane & 15);
    const int hi = lane >> 4;  // K sub-pair selector per ISA A-layout
    const float* arow = &smem[XS_OFF + row * STRIDE + 2 * hi];
    v8f acc = {0.f, 0.f, 0.f, 0.f, 0.f, 0.f, 0.f, 0.f};
    const v2f ones = {1.f, 1.f};
#pragma unroll
    for (int k = 0; k < DD; k += 4) {
      v2f a = *(const v2f*)(arow + k);
      acc = __builtin_amdgcn_wmma_f32_16x16x4_f32(false, a, false, ones,
                                                  (short)0, acc, false, false);
    }
    // Column 0 lives in lanes 0 (M=v) and 16 (M=v+8): write reciprocals.
    if ((lane & 15) == 0) {
      int base = wave * 16 + hi * 8;
#pragma unroll
      for (int v = 0; v < 8; ++v)
        smem[RS_OFF + base + v] = 1.0f / acc[v];
    }
  }
#else
  if (tid < TB) {
    const float* row = &smem[XS_OFF + tid * STRIDE];
    float s = 0.f;
#pragma unroll
    for (int d = 0; d < DD; d += 4) {
      float4 v = *(const float4*)(row + d);
      s += v.x + v.y + v.z + v.w;
    }
    smem[RS_OFF + tid] = 1.0f / s;
  }
#endif
  __syncthreads();

  // ---- main min-sum loop: each thread owns a 4x4 output sub-tile --------
  const int bb = (tid >> 4) << 2;  // 0..60
  const int cc = (tid & 15) << 2;  // 0..60
  float acc[4][4] = {{0.f}};

#pragma unroll 2
  for (int d = 0; d < DD; d += 4) {
    float4 xv[4], cv[4];
#pragma unroll
    for (int i = 0; i < 4; ++i)
      xv[i] = *(const float4*)&smem[XS_OFF + (bb + i) * STRIDE + d];
#pragma unroll
    for (int j = 0; j < 4; ++j)
      cv[j] = *(const float4*)&smem[CS_OFF + (cc + j) * STRIDE + d];
#pragma unroll
    for (int i = 0; i < 4; ++i) {
#pragma unroll
      for (int j = 0; j < 4; ++j) {
        acc[i][j] += fminf(xv[i].x, cv[j].x) + fminf(xv[i].y, cv[j].y) +
                     fminf(xv[i].z, cv[j].z) + fminf(xv[i].w, cv[j].w);
      }
    }
  }

  // ---- epilogue: normalize, vigilance mask, coalesced float4 stores -----
#pragma unroll
  for (int i = 0; i < 4; ++i) {
    const float inv = smem[RS_OFF + bb + i];
    float4 o;
    float m;
    m = acc[i][0] * inv; o.x = (m >= VIG) ? m : 0.0f;
    m = acc[i][1] * inv; o.y = (m >= VIG) ? m : 0.0f;
    m = acc[i][2] * inv; o.z = (m >= VIG) ? m : 0.0f;
    m = acc[i][3] * inv; o.w = (m >= VIG) ? m : 0.0f;
    *(float4*)(out + (size_t)(b0g + bb + i) * C + (c0g + cc)) = o;
  }
}

// One block per row: argmax with first-occurrence tie-break (jnp.argmax).
__global__ __launch_bounds__(256)
void fuzzyart_argmax_kernel(const float* __restrict__ scores,
                            float* __restrict__ idx_out, int C) {
  __shared__ float sval[256];
  __shared__ int sidx[256];
  const int row = blockIdx.x;
  const float* r = scores + (size_t)row * C;
  float best = -1.0f;
  int bidx = 0x7fffffff;
  for (int c = threadIdx.x; c < C; c += 256) {
    float v = r[c];
    if (v > best || (v == best && c < bidx)) { best = v; bidx = c; }
  }
  sval[threadIdx.x] = best;
  sidx[threadIdx.x] = bidx;
  __syncthreads();
  for (int s = 128; s > 0; s >>= 1) {
    if (threadIdx.x < s) {
      float v2 = sval[threadIdx.x + s];
      int i2 = sidx[threadIdx.x + s];
      if (v2 > sval[threadIdx.x] ||
          (v2 == sval[threadIdx.x] && i2 < sidx[threadIdx.x])) {
        sval[threadIdx.x] = v2;
        sidx[threadIdx.x] = i2;
      }
    }
    __syncthreads();
  }
  if (threadIdx.x == 0) idx_out[row] = (float)sidx[0];
}

extern "C" void kernel_launch(void* const* d_in, const int* in_sizes, int n_in,
                              void* d_out, int out_size, void* d_ws,
                              size_t ws_size, hipStream_t stream) {
  (void)n_in; (void)out_size; (void)d_ws; (void)ws_size;
  const float* x = (const float*)d_in[0];
  const float* cats = (const float*)d_in[1];
  const int B = in_sizes[0] / DD;  // 2048
  const int C = in_sizes[1] / DD;  // 512
  float* scores = (float*)d_out;
  float* idx = scores + (size_t)B * C;

  dim3 grid((unsigned)(B / TB), (unsigned)(C / TC));
  fuzzyart_match_kernel<<<grid, dim3(NTHREADS), SMEM_BYTES, stream>>>(
      x, cats, scores, B, C);
  fuzzyart_argmax_kernel<<<dim3((unsigned)B), dim3(256), 0, stream>>>(scores,
                                                                      idx, C);
}